// HGNLayer_38371237822894
// MI455X (gfx1250) — compile-verified
//
#include <hip/hip_runtime.h>
#include <math.h>

// Problem constants (match the reference).
#define R_REL 8
#define N_ENT 40000
#define F_IN  128
#define F_OUT 128
#define N_EDG 160000   // E per relation
#define TOT_EDGES (R_REL * N_EDG)  // 1,280,000

typedef __attribute__((ext_vector_type(2))) float v2f;
typedef __attribute__((ext_vector_type(8))) float v8f;

// ---------------------------------------------------------------------------
// Kernel 0: zero the `combined` accumulator (N_ENT x F_IN floats) in d_ws.
// ---------------------------------------------------------------------------
__global__ void hgn_zero_kernel(float4* __restrict__ ws, int n4) {
    int i = blockIdx.x * blockDim.x + threadIdx.x;
    if (i < n4) ws[i] = make_float4(0.f, 0.f, 0.f, 0.f);
}

// ---------------------------------------------------------------------------
// Kernel 1: alpha[r] = sigmoid( tanh(rel_mat @ W1 + b1) @ W2 ), r = 0..7.
// Also copies rel_mat (second tuple output) to the tail of d_out.
// One block of 256 threads; negligible cost.
// ---------------------------------------------------------------------------
__global__ void hgn_alpha_kernel(const float* __restrict__ rel,
                                 const float* __restrict__ w1,
                                 const float* __restrict__ b1,
                                 const float* __restrict__ w2,
                                 float* __restrict__ alpha,
                                 float* __restrict__ out_tail) {
    __shared__ float h[R_REL * F_OUT];
    int t = threadIdx.x;
    if (t < F_OUT) {
        float acc[R_REL];
        #pragma unroll
        for (int r = 0; r < R_REL; ++r) acc[r] = b1[t];
        for (int f = 0; f < F_IN; ++f) {
            float w = w1[f * F_OUT + t];
            #pragma unroll
            for (int r = 0; r < R_REL; ++r) acc[r] += rel[r * F_IN + f] * w;
        }
        #pragma unroll
        for (int r = 0; r < R_REL; ++r) h[r * F_OUT + t] = tanhf(acc[r]);
    }
    __syncthreads();
    if (t < R_REL) {
        float s = 0.f;
        for (int o = 0; o < F_OUT; ++o) s += h[t * F_OUT + o] * w2[o];
        alpha[t] = 1.0f / (1.0f + expf(-s));
    }
    // rel_mat passthrough -> d_out tail (R_REL*F_IN = 1024 floats)
    for (int i = t; i < R_REL * F_IN; i += blockDim.x) out_tail[i] = rel[i];
}

// ---------------------------------------------------------------------------
// Kernel 2: fused scatter-add. The per-relation scalar alpha[r] lets us fold
// the r-reduction into a single accumulator, so we never materialize the
// (R,N,F) supports tensor:
//   combined[src[r,e], :] += alpha[r]*val[r,e] * ent_mat[dst[r,e], :]
// One wave (32 lanes) per edge; each lane owns 4 consecutive floats
// (float4 gather from L2-resident ent_mat + 4 no-return f32 atomics).
// ---------------------------------------------------------------------------
__global__ void hgn_scatter_kernel(const float4* __restrict__ ent4,
                                   const int* __restrict__ esrc,
                                   const int* __restrict__ edst,
                                   const float* __restrict__ eval,
                                   const float* __restrict__ alpha,
                                   float* __restrict__ combined) {
    int wave = threadIdx.x >> 5;
    int lane = threadIdx.x & 31;
    int edge = blockIdx.x * 8 + wave;           // 8 edges per 256-thread block
    if (edge >= TOT_EDGES) return;
    int r = edge / N_EDG;
    int s = esrc[edge];
    int d = edst[edge];
    float scale = alpha[r] * eval[edge];
    float4 m = ent4[d * (F_IN / 4) + lane];     // 512B gather, L2-resident
    float* base = combined + (size_t)s * F_IN + lane * 4;
    atomicAdd(base + 0, scale * m.x);           // global_atomic_add_f32 (no-rtn)
    atomicAdd(base + 1, scale * m.y);
    atomicAdd(base + 2, scale * m.z);
    atomicAdd(base + 3, scale * m.w);
}

// ---------------------------------------------------------------------------
// Kernel 3: ent_output = combined(40000x128) @ weight_ent(128x128) in fp32
// using V_WMMA_F32_16X16X4_F32 (exact fp32 accumulate — matches reference
// dtype; this GEMM is compute-trivial so fp32 WMMA is the right choice).
//
// One wave per 16x16 output tile. Per ISA §7.12.2:
//   A 16x4 f32:  lanes 0-15 hold K=k0+0 (v.x), k0+1 (v.y); lanes 16-31 hold
//                K=k0+2, k0+3, row M = lane%16.
//   B 4x16 f32:  same K-per-lane-half pattern, col N = lane%16.
//   C/D 16x16:   VGPR j: lanes 0-15 -> M=j, lanes 16-31 -> M=8+j, N=lane%16.
// Block = 256 threads = 8 waves covering all 8 N-tiles of F_OUT; grid covers
// M exactly (2500*16 = 40000) so EXEC is all-1s as WMMA requires.
// ---------------------------------------------------------------------------
__global__ void hgn_gemm_wmma_kernel(const float* __restrict__ A,
                                     const float* __restrict__ W,
                                     float* __restrict__ Out) {
    int lane  = threadIdx.x & 31;
    int wave  = threadIdx.x >> 5;       // n-tile index 0..7
    int m0    = blockIdx.x * 16;
    int n0    = wave * 16;
    int l16   = lane & 15;
    int khalf = (lane >> 4) * 2;        // 0 for lanes 0-15, 2 for lanes 16-31

    const float* arow = A + (size_t)(m0 + l16) * F_IN + khalf;
    const float* wcol = W + (size_t)khalf * F_OUT + (n0 + l16);

    v8f c = {};
    #pragma unroll
    for (int k0 = 0; k0 < F_IN; k0 += 4) {
        v2f a, b;
        a.x = arow[k0];
        a.y = arow[k0 + 1];
        b.x = wcol[(size_t)k0 * F_OUT];
        b.y = wcol[(size_t)(k0 + 1) * F_OUT];
        // (neg_a, A, neg_b, B, c_mod, C, reuse_a, reuse_b)
        c = __builtin_amdgcn_wmma_f32_16x16x4_f32(false, a, false, b,
                                                  (short)0, c, false, false);
    }

    int orow = m0 + (lane >> 4) * 8;
    int col  = n0 + l16;
    #pragma unroll
    for (int j = 0; j < 8; ++j)
        Out[(size_t)(orow + j) * F_OUT + col] = c[j];
}

// ---------------------------------------------------------------------------
// Launch: zero -> alpha(+rel copy) -> scatter -> WMMA GEMM (same stream, so
// dependencies are satisfied by stream order). Workspace layout:
//   [0, N*F_IN) floats : combined accumulator
//   [N*F_IN, +8) floats: alpha
// ---------------------------------------------------------------------------
extern "C" void kernel_launch(void* const* d_in, const int* in_sizes, int n_in,
                              void* d_out, int out_size, void* d_ws, size_t ws_size,
                              hipStream_t stream) {
    const float* ent  = (const float*)d_in[0];
    const float* rel  = (const float*)d_in[1];
    const int*   esrc = (const int*)  d_in[2];
    const int*   edst = (const int*)  d_in[3];
    const float* ev   = (const float*)d_in[4];
    const float* went = (const float*)d_in[5];
    const float* pw1  = (const float*)d_in[6];
    const float* pb1  = (const float*)d_in[7];
    const float* pw2  = (const float*)d_in[8];

    float* out      = (float*)d_out;
    float* combined = (float*)d_ws;
    float* alpha    = combined + (size_t)N_ENT * F_IN;

    int n4 = N_ENT * F_IN / 4;  // 1,280,000 float4s
    hgn_zero_kernel<<<(n4 + 255) / 256, 256, 0, stream>>>((float4*)combined, n4);

    hgn_alpha_kernel<<<1, 256, 0, stream>>>(rel, pw1, pb1, pw2, alpha,
                                            out + (size_t)N_ENT * F_OUT);

    hgn_scatter_kernel<<<TOT_EDGES / 8, 256, 0, stream>>>(
        (const float4*)ent, esrc, edst, ev, alpha, combined);

    hgn_gemm_wmma_kernel<<<N_ENT / 16, 256, 0, stream>>>(combined, went, out);
}